// SAE_bias_pre_81363860455630
// MI455X (gfx1250) — compile-verified
//
#include <hip/hip_runtime.h>
#include <hip/hip_bf16.h>

// ---------------------------------------------------------------------------
// SAE forward: recon + dead_recon
//   BS=2048, D_IN=1024, D_SP=16384, K=32, THRESH=100
// Pipeline:
//   0) init workspace (hist/state/active)
//   1) W2 -> W2t transpose ; cvec[j] = dot(bias_pre, W1[j,:])
//   2) encoder GEMM scores = x @ W1^T - cvec  via v_wmma_f32_16x16x4_f32,
//      double-buffered LDS tiles filled with global_load_async_to_lds_b128
//   3) 3-level radix select over all scores -> global threshold key
//   4) mark active columns; derive dead flags
//   5) per-row radix select over dead columns -> per-row threshold key
//   6) sparse decode: recon = topk @ W2^T + bias ; dead_recon = dead_topk @ W2^T
// ---------------------------------------------------------------------------

typedef __attribute__((ext_vector_type(2))) float v2f;
typedef __attribute__((ext_vector_type(4))) float f4;
typedef __attribute__((ext_vector_type(8))) float v8f;
typedef __attribute__((ext_vector_type(4))) int   v4i;

#define BS    2048
#define D_IN  1024
#define D_SP  16384
#define KSEL  32
#define NTOT  (BS * D_SP)          // 33,554,432
#define GLOBAL_TK (KSEL * BS)      // 65536
#define ROW_TK    (2 * KSEL)       // 64

#define AS1 __attribute__((address_space(1)))
#define AS3 __attribute__((address_space(3)))

#if defined(__has_builtin)
#  if __has_builtin(__builtin_amdgcn_global_load_async_to_lds_b128)
#    define HAVE_ASYNC 1
#  endif
#endif
#ifndef HAVE_ASYNC
#  define HAVE_ASYNC 0
#endif

#if HAVE_ASYNC
#  if __has_builtin(__builtin_amdgcn_s_wait_asynccnt)
#    define WAIT_ASYNC0() __builtin_amdgcn_s_wait_asynccnt(0)
#  else
#    define WAIT_ASYNC0() asm volatile("s_wait_asynccnt 0x0" ::: "memory")
#  endif
__device__ __forceinline__ void async_cp_b128(const float* g, float* l) {
  __builtin_amdgcn_global_load_async_to_lds_b128(
      (AS1 v4i*)g, (AS3 v4i*)l, 0, 0);
}
#endif

// monotone float -> uint key: larger float <=> larger key
__device__ __forceinline__ unsigned fkey(float f) {
  unsigned u = __float_as_uint(f);
  return (u & 0x80000000u) ? ~u : (u | 0x80000000u);
}

// ---------------------------------------------------------------------------
// 0) init: zero 2048-bin histogram, select-state, active flags
// ---------------------------------------------------------------------------
__global__ void init_ws(unsigned* __restrict__ hist, unsigned* __restrict__ st,
                        unsigned* __restrict__ active) {
  int i = blockIdx.x * blockDim.x + threadIdx.x;
  if (i < 2048) hist[i] = 0u;
  if (i < D_SP) active[i] = 0u;
  if (i == 0) { st[0] = 0u; st[1] = (unsigned)GLOBAL_TK; st[2] = 0u; }
}

// ---------------------------------------------------------------------------
// 1a) W2 [D_IN, D_SP] -> W2t [D_SP, D_IN]
// ---------------------------------------------------------------------------
__global__ __launch_bounds__(256) void transpose_w2(const float* __restrict__ W2,
                                                    float* __restrict__ W2t) {
  __shared__ float t[32][33];
  const int jb = blockIdx.x * 32;          // along D_SP
  const int ib = blockIdx.y * 32;          // along D_IN
  const int lx = threadIdx.x & 31;
  const int ly = threadIdx.x >> 5;         // 8 rows per step
  for (int r = ly; r < 32; r += 8)
    t[r][lx] = W2[(size_t)(ib + r) * D_SP + jb + lx];
  __syncthreads();
  for (int r = ly; r < 32; r += 8)
    W2t[(size_t)(jb + r) * D_IN + ib + lx] = t[lx][r];
}

// ---------------------------------------------------------------------------
// 1b) cvec[j] = dot(bias_pre, W1[j,:])   (so  scores = x@W1^T - cvec)
// ---------------------------------------------------------------------------
__global__ __launch_bounds__(256) void col_corr(const float* __restrict__ W1,
                                                const float* __restrict__ bias,
                                                float* __restrict__ cvec) {
  int j = blockIdx.x * 256 + threadIdx.x;
  if (j >= D_SP) return;
  const float* r = W1 + (size_t)j * D_IN;
  float acc = 0.f;
  for (int k = 0; k < D_IN; k += 4) {
    f4 w = *(const f4*)(r + k);
    f4 b = *(const f4*)(bias + k);
    acc += w.x * b.x + w.y * b.y + w.z * b.z + w.w * b.w;
  }
  cvec[j] = acc;
}

// ---------------------------------------------------------------------------
// 2) encoder GEMM: scores[b,j] = sum_k x[b,k]*W1[j,k] - cvec[j]
//    64x128 tile per workgroup, 8 waves of 32x32, WMMA f32 16x16x4,
//    double-buffered LDS tiles via async global->LDS DMA
// ---------------------------------------------------------------------------
#define TM 64
#define TN 128
#define TK 32
#define LSTR 36   // padded LDS row stride (floats): 16B aligned, conflict-free
#define ASZ (TM * LSTR)
#define BSZ (TN * LSTR)

__global__ __launch_bounds__(256) void gemm_enc(const float* __restrict__ x,
                                                const float* __restrict__ W1,
                                                const float* __restrict__ cvec,
                                                float* __restrict__ scores) {
#if HAVE_ASYNC
  __shared__ float As[2 * ASZ];
  __shared__ float Bs[2 * BSZ];
#else
  __shared__ float As[ASZ];
  __shared__ float Bs[BSZ];
#endif

  const int tid  = threadIdx.x;
  const int lane = tid & 31;
  const int wave = tid >> 5;
  const int wm = (wave >> 2) * 32;        // 0 / 32
  const int wn = (wave & 3) * 32;         // 0..96
  const int bm = blockIdx.x * TM;
  const int bn = blockIdx.y * TN;

  v8f c00 = {}, c01 = {}, c10 = {}, c11 = {};

  const int ar  = tid >> 3;               // 0..31
  const int ac4 = (tid & 7) * 4;          // float col within TK

  const int koff = (lane >> 4) * 2;       // lanes 16-31 hold K+2,K+3
  const int lrow = lane & 15;

#if HAVE_ASYNC
  // ---- async double-buffered pipeline ----
  // issue chunk k0 into buffer `sel`
  auto issue = [&](int sel, int k0) {
    float* Ab = As + sel * ASZ;
    float* Bb = Bs + sel * BSZ;
    async_cp_b128(x + (size_t)(bm + ar) * D_IN + k0 + ac4,       &Ab[ar * LSTR + ac4]);
    async_cp_b128(x + (size_t)(bm + ar + 32) * D_IN + k0 + ac4,  &Ab[(ar + 32) * LSTR + ac4]);
#pragma unroll
    for (int i = 0; i < 4; ++i) {
      int r = ar + i * 32;
      async_cp_b128(W1 + (size_t)(bn + r) * D_IN + k0 + ac4, &Bb[r * LSTR + ac4]);
    }
  };

  issue(0, 0);
  WAIT_ASYNC0();
  __syncthreads();

  int sel = 0;
  for (int k0 = 0; k0 < D_IN; k0 += TK) {
    const int k1 = k0 + TK;
    if (k1 < D_IN) issue(sel ^ 1, k1);

    const float* Ab = As + sel * ASZ;
    const float* Bb = Bs + sel * BSZ;
#pragma unroll
    for (int kk = 0; kk < TK; kk += 4) {
      v2f a0 = *(const v2f*)&Ab[(wm + lrow) * LSTR + kk + koff];
      v2f a1 = *(const v2f*)&Ab[(wm + 16 + lrow) * LSTR + kk + koff];
      v2f b0 = *(const v2f*)&Bb[(wn + lrow) * LSTR + kk + koff];
      v2f b1 = *(const v2f*)&Bb[(wn + 16 + lrow) * LSTR + kk + koff];
      c00 = __builtin_amdgcn_wmma_f32_16x16x4_f32(false, a0, false, b0, (short)0, c00, false, false);
      c01 = __builtin_amdgcn_wmma_f32_16x16x4_f32(false, a0, false, b1, (short)0, c01, false, false);
      c10 = __builtin_amdgcn_wmma_f32_16x16x4_f32(false, a1, false, b0, (short)0, c10, false, false);
      c11 = __builtin_amdgcn_wmma_f32_16x16x4_f32(false, a1, false, b1, (short)0, c11, false, false);
    }
    if (k1 < D_IN) {
      WAIT_ASYNC0();
      __syncthreads();
    }
    sel ^= 1;
  }
#else
  // ---- synchronous fallback (single buffer) ----
  for (int k0 = 0; k0 < D_IN; k0 += TK) {
#pragma unroll
    for (int i = 0; i < 2; ++i) {
      int r = ar + i * 32;
      *(f4*)&As[r * LSTR + ac4] = *(const f4*)(x + (size_t)(bm + r) * D_IN + k0 + ac4);
    }
#pragma unroll
    for (int i = 0; i < 4; ++i) {
      int r = ar + i * 32;
      *(f4*)&Bs[r * LSTR + ac4] = *(const f4*)(W1 + (size_t)(bn + r) * D_IN + k0 + ac4);
    }
    __syncthreads();
#pragma unroll
    for (int kk = 0; kk < TK; kk += 4) {
      v2f a0 = *(const v2f*)&As[(wm + lrow) * LSTR + kk + koff];
      v2f a1 = *(const v2f*)&As[(wm + 16 + lrow) * LSTR + kk + koff];
      v2f b0 = *(const v2f*)&Bs[(wn + lrow) * LSTR + kk + koff];
      v2f b1 = *(const v2f*)&Bs[(wn + 16 + lrow) * LSTR + kk + koff];
      c00 = __builtin_amdgcn_wmma_f32_16x16x4_f32(false, a0, false, b0, (short)0, c00, false, false);
      c01 = __builtin_amdgcn_wmma_f32_16x16x4_f32(false, a0, false, b1, (short)0, c01, false, false);
      c10 = __builtin_amdgcn_wmma_f32_16x16x4_f32(false, a1, false, b0, (short)0, c10, false, false);
      c11 = __builtin_amdgcn_wmma_f32_16x16x4_f32(false, a1, false, b1, (short)0, c11, false, false);
    }
    __syncthreads();
  }
#endif

  // C layout: vgpr r, lanes 0-15 -> M = r, lanes 16-31 -> M = r + 8
  const int crow = (lane >> 4) * 8;
  const int ccol = lane & 15;
  const float cc0 = cvec[bn + wn + ccol];
  const float cc1 = cvec[bn + wn + ccol + 16];
#pragma unroll
  for (int r = 0; r < 8; ++r) {
    int m0 = bm + wm + crow + r;
    int n0 = bn + wn + ccol;
    scores[(size_t)m0 * D_SP + n0]             = c00[r] - cc0;
    scores[(size_t)m0 * D_SP + n0 + 16]        = c01[r] - cc1;
    scores[(size_t)(m0 + 16) * D_SP + n0]      = c10[r] - cc0;
    scores[(size_t)(m0 + 16) * D_SP + n0 + 16] = c11[r] - cc1;
  }
}

// ---------------------------------------------------------------------------
// 3) radix-select histogram + scan (3 levels: bits [31:21], [20:10], [9:0])
// ---------------------------------------------------------------------------
__device__ __forceinline__ void level_params(int level, unsigned& mask,
                                             int& shift, int& bins) {
  if (level == 0)      { mask = 0u;          shift = 21; bins = 2048; }
  else if (level == 1) { mask = 0xFFE00000u; shift = 10; bins = 2048; }
  else                 { mask = 0xFFFFFC00u; shift = 0;  bins = 1024; }
}

__global__ __launch_bounds__(256) void hist_pass(const float* __restrict__ sc,
                                                 int n,
                                                 const unsigned* __restrict__ st,
                                                 unsigned* __restrict__ hist,
                                                 int level) {
  __shared__ unsigned h[2048];
  for (int i = threadIdx.x; i < 2048; i += 256) h[i] = 0u;
  __syncthreads();
  unsigned mask; int shift, bins;
  level_params(level, mask, shift, bins);
  const unsigned prefix = st[0];
  for (int i = blockIdx.x * 256 + threadIdx.x; i < n; i += gridDim.x * 256) {
    unsigned k = fkey(sc[i]);
    if ((k & mask) == prefix)
      atomicAdd(&h[(k >> shift) & (unsigned)(bins - 1)], 1u);
  }
  __syncthreads();
  for (int i = threadIdx.x; i < bins; i += 256) {
    unsigned v = h[i];
    if (v) atomicAdd(&hist[i], v);
  }
}

__global__ __launch_bounds__(256) void scan_pass(unsigned* __restrict__ st,
                                                 unsigned* __restrict__ hist,
                                                 int level) {
  if (threadIdx.x == 0) {
    unsigned mask; int shift, bins;
    level_params(level, mask, shift, bins);
    unsigned T = st[1];
    unsigned cum = 0;
    int found = -1;
    for (int b = bins - 1; b >= 0; --b) {
      unsigned c = hist[b];
      if (cum + c >= T) { found = b; st[1] = T - cum; break; }
      cum += c;
    }
    if (found < 0) { found = 0; st[1] = 0u; }
    st[0] |= ((unsigned)found) << shift;
    if (level == 2) st[2] = st[0];      // final threshold key
  }
  __syncthreads();
  for (int i = threadIdx.x; i < 2048; i += blockDim.x) hist[i] = 0u;
}

// ---------------------------------------------------------------------------
// 4) active / dead flags
// ---------------------------------------------------------------------------
__global__ __launch_bounds__(256) void mark_active(const float* __restrict__ sc,
                                                   int n,
                                                   const unsigned* __restrict__ st,
                                                   unsigned* __restrict__ active) {
  const unsigned th = st[2];
  for (int i = blockIdx.x * 256 + threadIdx.x; i < n; i += gridDim.x * 256) {
    if (fkey(sc[i]) >= th) active[i & (D_SP - 1)] = 1u;   // benign race: same value
  }
}

__global__ void dead_flags(const unsigned* __restrict__ active,
                           const float* __restrict__ la,
                           unsigned* __restrict__ dead) {
  int j = blockIdx.x * blockDim.x + threadIdx.x;
  if (j < D_SP)
    dead[j] = (!active[j] && (la[j] + 1.0f) > 100.0f) ? 1u : 0u;
}

// ---------------------------------------------------------------------------
// 5) per-row top-64 over dead columns -> per-row threshold key
// ---------------------------------------------------------------------------
__global__ __launch_bounds__(256) void row_select(const float* __restrict__ sc,
                                                  const unsigned* __restrict__ dead,
                                                  unsigned* __restrict__ rowkey) {
  __shared__ unsigned h[2048];
  __shared__ unsigned s_prefix, s_target, s_done;
  const int b = blockIdx.x;
  const float* row = sc + (size_t)b * D_SP;
  if (threadIdx.x == 0) { s_prefix = 0u; s_target = (unsigned)ROW_TK; s_done = 0u; }

  for (int lvl = 0; lvl < 3; ++lvl) {
    __syncthreads();
    for (int i = threadIdx.x; i < 2048; i += 256) h[i] = 0u;
    __syncthreads();
    unsigned mask; int shift, bins;
    level_params(lvl, mask, shift, bins);
    if (!s_done) {
      const unsigned prefix = s_prefix;
      for (int j = threadIdx.x; j < D_SP; j += 256) {
        if (dead[j]) {
          unsigned k = fkey(row[j]);
          if ((k & mask) == prefix)
            atomicAdd(&h[(k >> shift) & (unsigned)(bins - 1)], 1u);
        }
      }
    }
    __syncthreads();
    if (threadIdx.x == 0 && !s_done) {
      unsigned T = s_target, cum = 0;
      int found = -1;
      for (int bin = bins - 1; bin >= 0; --bin) {
        unsigned c = h[bin];
        if (cum + c >= T) { found = bin; s_target = T - cum; break; }
        cum += c;
      }
      if (found < 0) { s_done = 1u; }                  // < 64 dead: take all
      else s_prefix |= ((unsigned)found) << shift;
    }
  }
  __syncthreads();
  if (threadIdx.x == 0) rowkey[b] = s_done ? 0u : s_prefix;
}

// ---------------------------------------------------------------------------
// 6) sparse decode: one workgroup per batch row
// ---------------------------------------------------------------------------
#define CAP 2048
__global__ __launch_bounds__(256) void decode(const float* __restrict__ sc,
                                              const unsigned* __restrict__ st,
                                              const unsigned* __restrict__ dead,
                                              const unsigned* __restrict__ rowkey,
                                              const float* __restrict__ W2t,
                                              const float* __restrict__ bias,
                                              float* __restrict__ out) {
  __shared__ float    lv1[CAP]; __shared__ unsigned li1[CAP];
  __shared__ float    lv2[CAP]; __shared__ unsigned li2[CAP];
  __shared__ unsigned c1, c2;
  const int b = blockIdx.x;
  const int tid = threadIdx.x;
  if (tid == 0) { c1 = 0u; c2 = 0u; }
  __syncthreads();

  const unsigned gth = st[2];
  const unsigned rk  = rowkey[b];
  const float* row = sc + (size_t)b * D_SP;

  for (int j = tid; j < D_SP; j += 256) {
    float s = row[j];
    unsigned k = fkey(s);
    if (k >= gth) {
      unsigned p = atomicAdd(&c1, 1u);
      if (p < CAP) { li1[p] = (unsigned)j; lv1[p] = s; }
    }
    if (dead[j] && k >= rk) {
      unsigned p = atomicAdd(&c2, 1u);
      if (p < CAP) { li2[p] = (unsigned)j; lv2[p] = s; }
    }
  }
  __syncthreads();

  unsigned n1 = c1 < (unsigned)CAP ? c1 : (unsigned)CAP;
  unsigned n2 = c2 < (unsigned)CAP ? c2 : (unsigned)CAP;

  f4 a1 = {0.f, 0.f, 0.f, 0.f};
  f4 a2 = {0.f, 0.f, 0.f, 0.f};
  const int col = tid * 4;
  for (unsigned e = 0; e < n1; ++e) {
    float v = lv1[e];
    f4 w = *(const f4*)(W2t + (size_t)li1[e] * D_IN + col);
    a1 = a1 + v * w;
  }
  for (unsigned e = 0; e < n2; ++e) {
    float v = lv2[e];
    f4 w = *(const f4*)(W2t + (size_t)li2[e] * D_IN + col);
    a2 = a2 + v * w;
  }
  f4 bp = *(const f4*)(bias + col);
  *(f4*)(out + (size_t)b * D_IN + col) = a1 + bp;
  *(f4*)(out + (size_t)BS * D_IN + (size_t)b * D_IN + col) = a2;
}

// ---------------------------------------------------------------------------
// launch
// ---------------------------------------------------------------------------
extern "C" void kernel_launch(void* const* d_in, const int* in_sizes, int n_in,
                              void* d_out, int out_size, void* d_ws, size_t ws_size,
                              hipStream_t stream) {
  (void)in_sizes; (void)n_in; (void)out_size; (void)ws_size;
  const float* x    = (const float*)d_in[0];
  const float* W1   = (const float*)d_in[1];
  const float* W2   = (const float*)d_in[2];
  const float* bias = (const float*)d_in[3];
  const float* la   = (const float*)d_in[4];
  float* out        = (float*)d_out;

  // workspace layout
  float*    scores  = (float*)d_ws;                         // 128 MB
  float*    W2t     = scores + (size_t)BS * D_SP;           //  64 MB
  unsigned* hist    = (unsigned*)(W2t + (size_t)D_SP * D_IN);
  unsigned* st      = hist + 2048;
  unsigned* active  = st + 16;
  unsigned* dead    = active + D_SP;
  unsigned* rowkey  = dead + D_SP;
  float*    cvec    = (float*)(rowkey + BS);

  init_ws<<<64, 256, 0, stream>>>(hist, st, active);
  transpose_w2<<<dim3(D_SP / 32, D_IN / 32), 256, 0, stream>>>(W2, W2t);
  col_corr<<<D_SP / 256, 256, 0, stream>>>(W1, bias, cvec);
  gemm_enc<<<dim3(BS / TM, D_SP / TN), 256, 0, stream>>>(x, W1, cvec, scores);

  for (int lvl = 0; lvl < 3; ++lvl) {
    hist_pass<<<2048, 256, 0, stream>>>(scores, NTOT, st, hist, lvl);
    scan_pass<<<1, 256, 0, stream>>>(st, hist, lvl);
  }

  mark_active<<<2048, 256, 0, stream>>>(scores, NTOT, st, active);
  dead_flags<<<D_SP / 256, 256, 0, stream>>>(active, la, dead);
  row_select<<<BS, 256, 0, stream>>>(scores, dead, rowkey);
  decode<<<BS, 256, 0, stream>>>(scores, st, dead, rowkey, W2t, bias, out);
}